// MinibatchDiscrimination_2860448219172
// MI455X (gfx1250) — compile-verified
//
#include <hip/hip_runtime.h>
#include <hip/hip_bf16.h>

typedef __attribute__((ext_vector_type(2))) float v2f;
typedef __attribute__((ext_vector_type(8))) float v8f;

#define NN 512
#define FF 512
#define OO 128
#define KK 8
#define OK 1024   // O*K, columns of projected matrix
#define OUTW 640  // F + O

// -------------------------------------------------------------------------
// Kernel 1: m[512][1024] = x[512][512] @ Tm[512][1024], exact fp32 via
// V_WMMA_F32_16X16X4_F32. One wave per 16x16 output tile, 8 waves/block.
// -------------------------------------------------------------------------
__global__ __launch_bounds__(256) void mbd_gemm_wmma(
    const float* __restrict__ x, const float* __restrict__ Tm,
    float* __restrict__ m) {
  const int lane  = threadIdx.x & 31;
  const int wave  = threadIdx.x >> 5;
  const int tile  = blockIdx.x * 8 + wave;   // 2048 tiles total
  const int tiles_n = OK / 16;               // 64 tiles across columns
  const int tm = (tile / tiles_n) * 16;      // row base of tile
  const int tn = (tile % tiles_n) * 16;      // col base of tile
  const int lhalf = lane >> 4;               // 0: K pair {0,1}, 1: K pair {2,3}
  const int l15   = lane & 15;

  v8f acc = {};
  // A-frag: lane holds x[tm+l15][k + lhalf*2 + {0,1}] -> contiguous float2
  const float* aptr = x + (tm + l15) * FF + lhalf * 2;
  // B-frag: lane holds Tm[k + lhalf*2 + {0,1}][tn+l15]
  const float* bptr = Tm + (lhalf * 2) * OK + tn + l15;

#pragma unroll 4
  for (int k = 0; k < FF; k += 4) {
    v2f a = *(const v2f*)(aptr + k);
    v2f b;
    b.x = bptr[(size_t)k * OK];
    b.y = bptr[(size_t)k * OK + OK];
    acc = __builtin_amdgcn_wmma_f32_16x16x4_f32(
        /*neg_a=*/false, a, /*neg_b=*/false, b,
        /*c_mod=*/(short)0, acc, /*reuse_a=*/false, /*reuse_b=*/false);
  }

  // D layout: VGPR j -> D[M = j + 8*lhalf][N = l15]
  float* dptr = m + (size_t)(tm + lhalf * 8) * OK + tn + l15;
#pragma unroll
  for (int j = 0; j < 8; ++j) dptr[(size_t)j * OK] = acc[j];
}

// -------------------------------------------------------------------------
// Kernel 2: copy x into out[:, 0:512] (row stride 640), vectorized float4.
// -------------------------------------------------------------------------
__global__ __launch_bounds__(256) void mbd_copy_x(
    const float* __restrict__ x, float* __restrict__ out) {
  int idx = blockIdx.x * 256 + threadIdx.x;  // 512 rows * 128 float4/row
  int r = idx >> 7;
  int c = idx & 127;
  ((float4*)(out + (size_t)r * OUTW))[c] = ((const float4*)(x + (size_t)r * FF))[c];
}

// -------------------------------------------------------------------------
// Kernel 3: pairwise L1 over K + exp-sum.
// Block = (o, i-chunk): stage m[:, o, :] (16 KB) in LDS once, then each
// thread owns one row i; inner j-loop reads LDS as a wave-wide broadcast
// (ds_load_b128), accumulates exp(-L1) with v_exp_f32.
// -------------------------------------------------------------------------
__global__ __launch_bounds__(128) void mbd_pairwise(
    const float* __restrict__ m, float* __restrict__ out) {
  __shared__ float sm[NN * KK];              // 16 KB: m[:, o, :]
  const int o     = blockIdx.x;              // 0..127
  const int chunk = blockIdx.y;              // 0..3

  // Cooperative stage: 128 threads * 4 rows each, two float4 per row.
  for (int r = threadIdx.x; r < NN; r += 128) {
    const float4* src = (const float4*)(m + (size_t)r * OK + o * KK);
    float4* dst = (float4*)(sm + r * KK);
    dst[0] = src[0];
    dst[1] = src[1];
  }
  __syncthreads();

  const int i = chunk * 128 + threadIdx.x;
  float4 mi0 = *(const float4*)(sm + i * KK);
  float4 mi1 = *(const float4*)(sm + i * KK + 4);

  float acc = 0.0f;
  for (int j = 0; j < NN; ++j) {
    float4 a0 = *(const float4*)(sm + j * KK);      // broadcast read
    float4 a1 = *(const float4*)(sm + j * KK + 4);
    float d = fabsf(a0.x - mi0.x) + fabsf(a0.y - mi0.y) +
              fabsf(a0.z - mi0.z) + fabsf(a0.w - mi0.w) +
              fabsf(a1.x - mi1.x) + fabsf(a1.y - mi1.y) +
              fabsf(a1.z - mi1.z) + fabsf(a1.w - mi1.w);
    acc += __expf(-d);                               // includes self: exp(0)=1
  }
  out[(size_t)i * OUTW + FF + o] = acc;
}

// -------------------------------------------------------------------------
extern "C" void kernel_launch(void* const* d_in, const int* in_sizes, int n_in,
                              void* d_out, int out_size, void* d_ws, size_t ws_size,
                              hipStream_t stream) {
  const float* x  = (const float*)d_in[0];   // [512, 512]
  const float* T  = (const float*)d_in[1];   // [512, 128, 8] == [512, 1024] flat
  float* out = (float*)d_out;                // [512, 640]
  float* m   = (float*)d_ws;                 // scratch: [512, 1024] = 2 MB

  mbd_gemm_wmma<<<256, 256, 0, stream>>>(x, T, m);
  mbd_copy_x<<<256, 256, 0, stream>>>(x, out);
  dim3 pg(OO, NN / 128);                     // 128 x 4 blocks
  mbd_pairwise<<<pg, 128, 0, stream>>>(m, out);
}